// HAN_1228360647040
// MI455X (gfx1250) — compile-verified
//
#include <hip/hip_runtime.h>
#include <hip/hip_bf16.h>

typedef float v2f __attribute__((ext_vector_type(2)));
typedef float v8f __attribute__((ext_vector_type(8)));

#define NEG_SLOPE 0.2f

// ---------------------------------------------------------------------------
// helpers
// ---------------------------------------------------------------------------
__device__ __forceinline__ void atomic_add_f32(float* p, float v) {
    __hip_atomic_fetch_add(p, v, __ATOMIC_RELAXED, __HIP_MEMORY_SCOPE_AGENT);
}

// fp32 atomic max via signed/unsigned integer punning (valid for non-NaN).
// Requires destination initialized to -inf (0xFF800000).
__device__ __forceinline__ void atomic_max_f32(float* p, float v) {
    if (v >= 0.0f) {
        atomicMax((int*)p, __float_as_int(v));
    } else {
        atomicMin((unsigned int*)p, __float_as_uint(v));
    }
}

// ---------------------------------------------------------------------------
// Strip-mined fp32 WMMA GEMM:  C[N,128] (+)= A[N,KTOT] @ B[KTOT,128]
// block = 128 threads (4 waves); each wave owns a 16x128 output strip
// (8 accumulator tiles), so the A fragment is loaded once and reused by
// 8 consecutive v_wmma_f32_16x16x4_f32. B is staged through LDS in 32KB
// 64x128 K-chunks with coalesced float4 loads.
// grid.x = ceil(ceil(N/16)/4)
// ---------------------------------------------------------------------------
template <int KTOT, bool ACC>
__global__ void wmma_gemm_kernel(const float* __restrict__ A,
                                 const float* __restrict__ B,
                                 float* __restrict__ C, int N) {
    __shared__ float sB[64 * 128];              // 32 KB K-chunk of B

    const int tid   = threadIdx.x;
    const int wave  = tid >> 5;
    const int lane  = tid & 31;
    const int tiles = (N + 15) >> 4;
    int mt = blockIdx.x * 4 + wave;
    if (mt >= tiles) mt = tiles - 1;            // clamp (dup work) so all waves
                                                // reach barriers; EXEC stays 1s
    const int l16   = lane & 15;
    const int khalf = lane >> 4;                // 0 or 1

    const float* ar = A + (size_t)(mt * 16 + l16) * KTOT + 2 * khalf;

    v8f acc[8];
    if (ACC) {
#pragma unroll
        for (int t = 0; t < 8; ++t)
#pragma unroll
            for (int v = 0; v < 8; ++v)
                acc[t][v] = C[(size_t)(mt * 16 + v + 8 * khalf) * 128 + t * 16 + l16];
    } else {
#pragma unroll
        for (int t = 0; t < 8; ++t) acc[t] = (v8f){};
    }

    for (int kc = 0; kc < KTOT; kc += 64) {
        __syncthreads();
        // cooperative, fully-coalesced load of B[kc..kc+64) x 128 into LDS
#pragma unroll
        for (int i = 0; i < 16; ++i) {
            int flat = i * 128 + tid;           // float4 index within chunk
            *(float4*)(sB + flat * 4) =
                *(const float4*)(B + (size_t)kc * 128 + flat * 4);
        }
        __syncthreads();

#pragma unroll 4
        for (int kk = 0; kk < 64; kk += 4) {
            v2f a = *(const v2f*)(ar + kc + kk); // A: rows L%16, K=2*khalf+{0,1}
            const float* bb = sB + (kk + 2 * khalf) * 128 + l16;
#pragma unroll
            for (int t = 0; t < 8; ++t) {
                v2f b;
                b.x = bb[t * 16];                // B: K = kk + 2*khalf,   N = t*16+l16
                b.y = bb[t * 16 + 128];          //    K = kk + 2*khalf+1
                acc[t] = __builtin_amdgcn_wmma_f32_16x16x4_f32(
                    false, a, false, b, (short)0, acc[t], false, false);
            }
        }
    }

#pragma unroll
    for (int t = 0; t < 8; ++t)
#pragma unroll
        for (int v = 0; v < 8; ++v)
            C[(size_t)(mt * 16 + v + 8 * khalf) * 128 + t * 16 + l16] = acc[t][v];
}

// ---------------------------------------------------------------------------
// Per-node head logits: s[n,h] = <h_row, a_src[h]>, t[n,h] = <h_row, a_dst[h]>
// one thread per (n, head)
// ---------------------------------------------------------------------------
__global__ void st_kernel(const float* __restrict__ h,
                          const float* __restrict__ asrc,
                          const float* __restrict__ adst,
                          float* __restrict__ s, float* __restrict__ t, int NH) {
    int i = blockIdx.x * blockDim.x + threadIdx.x;
    if (i >= NH) return;
    int n = i >> 3, hh = i & 7;
    const float* hp = h + (size_t)n * 128 + hh * 16;
    const float* as = asrc + hh * 16;
    const float* ad = adst + hh * 16;
    float ss = 0.f, tt = 0.f;
#pragma unroll
    for (int d = 0; d < 16; ++d) {
        float v = hp[d];
        ss += v * as[d];
        tt += v * ad[d];
    }
    s[i] = ss;
    t[i] = tt;
}

__global__ void init_softmax_kernel(float* __restrict__ mmax,
                                    float* __restrict__ den, int NH) {
    int i = blockIdx.x * blockDim.x + threadIdx.x;
    if (i >= NH) return;
    mmax[i] = -__builtin_huge_valf();
    den[i]  = 0.0f;
}

// ---------------------------------------------------------------------------
// Edge phase A: e = leaky_relu(s[src]+t[dst]); store; atomic segment-max
// one thread per edge (8 heads each)
// ---------------------------------------------------------------------------
__global__ void edge_logits_kernel(const int* __restrict__ ei,
                                   const float* __restrict__ s,
                                   const float* __restrict__ t,
                                   float* __restrict__ ebuf,
                                   float* __restrict__ mmax, int E) {
    int e = blockIdx.x * blockDim.x + threadIdx.x;
    if (e >= E) return;
    int src = ei[e];
    int dst = ei[E + e];
    const float* sp = s + (size_t)src * 8;
    const float* tp = t + (size_t)dst * 8;
    float* ep = ebuf + (size_t)e * 8;
    float* mp = mmax + (size_t)dst * 8;
#pragma unroll
    for (int hh = 0; hh < 8; ++hh) {
        float v = sp[hh] + tp[hh];
        v = v > 0.f ? v : NEG_SLOPE * v;
        ep[hh] = v;
        atomic_max_f32(&mp[hh], v);
    }
}

// Edge phase B: ex = exp(e - mmax[dst]); store; atomic denom sum
__global__ void edge_exp_kernel(const int* __restrict__ ei,
                                float* __restrict__ ebuf,
                                const float* __restrict__ mmax,
                                float* __restrict__ den, int E) {
    int e = blockIdx.x * blockDim.x + threadIdx.x;
    if (e >= E) return;
    int dst = ei[E + e];
    float* ep = ebuf + (size_t)e * 8;
    const float* mp = mmax + (size_t)dst * 8;
    float* dp = den + (size_t)dst * 8;
#pragma unroll
    for (int hh = 0; hh < 8; ++hh) {
        float ex = __expf(ep[hh] - mp[hh]);
        ep[hh] = ex;
        atomic_add_f32(&dp[hh], ex);
    }
}

// ---------------------------------------------------------------------------
// Edge phase C: agg[dst] += ex * h[src]
// one WAVE per edge: 32 lanes cover the 128-float row with one coalesced
// 512B float4 gather, then 4 global_atomic_add_f32 per lane.
// ---------------------------------------------------------------------------
__global__ void edge_agg_kernel(const int* __restrict__ ei,
                                const float* __restrict__ ebuf,
                                const float* __restrict__ h,
                                float* __restrict__ agg, int E) {
    int gid  = blockIdx.x * blockDim.x + threadIdx.x;
    int e    = gid >> 5;                        // one wave per edge
    int lane = gid & 31;
    if (e >= E) return;
    int src = ei[e];
    int dst = ei[E + e];
    float ex = ebuf[(size_t)e * 8 + (lane >> 2)];        // head = lane/4
    const float4 hv = *(const float4*)(h + (size_t)src * 128 + lane * 4);
    float* ap = agg + (size_t)dst * 128 + lane * 4;
    atomic_add_f32(ap + 0, ex * hv.x);
    atomic_add_f32(ap + 1, ex * hv.y);
    atomic_add_f32(ap + 2, ex * hv.z);
    atomic_add_f32(ap + 3, ex * hv.w);
}

// Normalize: agg[n, h*16+d] *= (den>0 ? 1/max(den,1e-16) : 0)
__global__ void normalize_kernel(float* __restrict__ agg,
                                 const float* __restrict__ den, int NH) {
    int i = blockIdx.x * blockDim.x + threadIdx.x;
    if (i >= NH) return;
    int n = i >> 3, hh = i & 7;
    float d = den[i];
    float scale = (d > 0.f) ? (1.0f / fmaxf(d, 1e-16f)) : 0.0f;
    float* ap = agg + (size_t)n * 128 + hh * 16;
#pragma unroll
    for (int dd = 0; dd < 16; ++dd) ap[dd] *= scale;
}

// Final epilogue: out = relu(out + bias)
__global__ void bias_relu_kernel(float* __restrict__ out,
                                 const float* __restrict__ b, int total) {
    int i = blockIdx.x * blockDim.x + threadIdx.x;
    if (i >= total) return;
    float v = out[i] + b[i & 127];
    out[i] = v > 0.f ? v : 0.f;
}

// ---------------------------------------------------------------------------
// launch
// ---------------------------------------------------------------------------
extern "C" void kernel_launch(void* const* d_in, const int* in_sizes, int n_in,
                              void* d_out, int out_size, void* d_ws, size_t ws_size,
                              hipStream_t stream) {
    const float* x     = (const float*)d_in[0];   // [N, 256]
    const float* W     = (const float*)d_in[1];   // [8, 256, 128]
    const float* a_src = (const float*)d_in[2];   // [8, 8, 16]
    const float* a_dst = (const float*)d_in[3];   // [8, 8, 16]
    const float* lin_W = (const float*)d_in[4];   // [1024, 128]
    const float* lin_b = (const float*)d_in[5];   // [128]
    const int*   ei    = (const int*)d_in[6];     // [8, 2, E] (int32: JAX x64 off)

    const int M  = 8;
    const int N  = in_sizes[0] / 256;
    const int E  = in_sizes[6] / (2 * M);
    const int NH = N * 8;

    float* out = (float*)d_out;                   // [N, 128], used as accumulator

    // workspace layout (floats)
    float* wsf     = (float*)d_ws;
    float* hbuf    = wsf;                         // N*128
    float* aggbuf  = hbuf    + (size_t)N * 128;   // N*128
    float* ebuf    = aggbuf  + (size_t)N * 128;   // E*8
    float* sbuf    = ebuf    + (size_t)E * 8;     // N*8
    float* tbuf    = sbuf    + (size_t)N * 8;     // N*8
    float* mmaxbuf = tbuf    + (size_t)N * 8;     // N*8
    float* denbuf  = mmaxbuf + (size_t)N * 8;     // N*8

    const int tiles = (N + 15) / 16;
    const int gblocks = (tiles + 3) / 4;
    const int TB = 256;
    const int nh_blocks = (NH + TB - 1) / TB;
    const int e_blocks  = (E + TB - 1) / TB;
    const int ew_blocks = ((E * 32) + TB - 1) / TB;   // wave-per-edge

    // zero the output accumulator
    hipMemsetAsync(out, 0, (size_t)N * 128 * sizeof(float), stream);

    for (int m = 0; m < M; ++m) {
        const float* Wm   = W     + (size_t)m * 256 * 128;
        const float* asm_ = a_src + (size_t)m * 8 * 16;
        const float* adm  = a_dst + (size_t)m * 8 * 16;
        const float* Lm   = lin_W + (size_t)m * 128 * 128;
        const int*   eim  = ei    + (size_t)m * 2 * E;

        wmma_gemm_kernel<256, false><<<gblocks, 128, 0, stream>>>(x, Wm, hbuf, N);
        st_kernel<<<nh_blocks, TB, 0, stream>>>(hbuf, asm_, adm, sbuf, tbuf, NH);
        init_softmax_kernel<<<nh_blocks, TB, 0, stream>>>(mmaxbuf, denbuf, NH);
        hipMemsetAsync(aggbuf, 0, (size_t)N * 128 * sizeof(float), stream);

        edge_logits_kernel<<<e_blocks, TB, 0, stream>>>(eim, sbuf, tbuf, ebuf, mmaxbuf, E);
        edge_exp_kernel<<<e_blocks, TB, 0, stream>>>(eim, ebuf, mmaxbuf, denbuf, E);
        edge_agg_kernel<<<ew_blocks, TB, 0, stream>>>(eim, ebuf, hbuf, aggbuf, E);

        normalize_kernel<<<nh_blocks, TB, 0, stream>>>(aggbuf, denbuf, NH);
        wmma_gemm_kernel<128, true><<<gblocks, 128, 0, stream>>>(aggbuf, Lm, out, N);
    }

    bias_relu_kernel<<<(N * 128 + TB - 1) / TB, TB, 0, stream>>>(out, lin_b, N * 128);
}